// InteractionBlock_14276471292010
// MI455X (gfx1250) — compile-verified
//
#include <hip/hip_runtime.h>
#include <math.h>

typedef float v2f __attribute__((ext_vector_type(2)));
typedef float v8f __attribute__((ext_vector_type(8)));

#define CC 64

// ---------------------------------------------------------------------------
// Wave-level fp32 GEMM: D(16 x 64) += A(16 x K) * B(K x 64) via
// V_WMMA_F32_16X16X4_F32.  loadA2(kb, k0) returns the lane's A-fragment pair
// (elements k0 and k0+1 of the lane's row), where k0 = kb + 2*(lane>>4).
// B is a row-major (K x 64) weight matrix (L2-resident, shared by all waves).
// ---------------------------------------------------------------------------
template <typename F>
__device__ __forceinline__ void run_gemm16x64(F loadA2, const float* __restrict__ B,
                                              int K, int lane, v8f c[4]) {
  const int ncol = lane & 15;
  const int kh   = (lane >> 4) * 2;   // 0 or 2
  for (int kb = 0; kb < K; kb += 4) {
    const int k0 = kb + kh;
    v2f a = loadA2(kb, k0);
#pragma unroll
    for (int nt = 0; nt < 4; ++nt) {
      v2f b;
      b.x = B[(size_t)k0 * CC + nt * 16 + ncol];
      b.y = B[(size_t)(k0 + 1) * CC + nt * 16 + ncol];
      // (neg_a, A, neg_b, B, c_mod, C, reuse_a, reuse_b)
      c[nt] = __builtin_amdgcn_wmma_f32_16x16x4_f32(false, a, false, b,
                                                    (short)0, c[nt], false, false);
    }
  }
}

// ---------------------------------------------------------------------------
// Node-side kernel.  gridDim.y selects the variant:
//   0      : y0    = x0 @ W_lin1_0 / 8                   (K=64)
//   1..3   : y1[m] = x1[:,:,m] @ W_lin1_1 / 8            (K=64)
//   4      : sc0   = (x0 (x) attrs) @ W_sc0 / 16  -> out (K=256)
//   5..7   : sc1[m]= (x1m (x) attrs) @ W_sc1 / 16 -> out (K=256)
// One wave per 16-node tile; 4 waves per block.  All branches wave-uniform,
// so EXEC is all-ones at every WMMA.
// ---------------------------------------------------------------------------
__global__ void node_gemm_kernel(const float* __restrict__ nf,
                                 const float* __restrict__ attrs,
                                 const float* __restrict__ Wl10,
                                 const float* __restrict__ Wl11,
                                 const float* __restrict__ Wsc0,
                                 const float* __restrict__ Wsc1,
                                 float* __restrict__ y0,
                                 float* __restrict__ y1,
                                 float* __restrict__ out,
                                 int N, int nTiles) {
  const int lane = threadIdx.x & 31;
  const int wave = threadIdx.x >> 5;
  const int tile = blockIdx.x * (blockDim.x >> 5) + wave;
  if (tile >= nTiles) return;               // wave-uniform
  const int variant = blockIdx.y;
  const int kh = (lane >> 4) * 2;

  int arow = tile * 16 + (lane & 15);
  if (arow >= N) arow = N - 1;              // clamp A loads for tail tiles
  const size_t an = (size_t)arow;

  v8f c[4] = {};

  if (variant == 0) {
    // contiguous, 8B-aligned pair -> single b64 load
    run_gemm16x64([&](int kb, int k0) {
      return *(const v2f*)(nf + an * 256 + k0);
    }, Wl10, 64, lane, c);
  } else if (variant <= 3) {
    const int m = variant - 1;
    run_gemm16x64([&](int kb, int k0) {
      v2f a;
      a.x = nf[an * 256 + 64 + k0 * 3 + m];
      a.y = nf[an * 256 + 64 + (k0 + 1) * 3 + m];
      return a;
    }, Wl11, 64, lane, c);
  } else if (variant == 4) {
    // k = u*4 + a ; both elements of the lane's pair share u = kb>>2,
    // attrs factors are loop-invariant per lane.
    const float atA = attrs[an * 4 + kh];
    const float atB = attrs[an * 4 + kh + 1];
    run_gemm16x64([&](int kb, int k0) {
      const float xv = nf[an * 256 + (kb >> 2)];
      v2f a; a.x = xv * atA; a.y = xv * atB;
      return a;
    }, Wsc0, 256, lane, c);
  } else {
    const int m = variant - 5;
    const float atA = attrs[an * 4 + kh];
    const float atB = attrs[an * 4 + kh + 1];
    run_gemm16x64([&](int kb, int k0) {
      const float xv = nf[an * 256 + 64 + (kb >> 2) * 3 + m];
      v2f a; a.x = xv * atA; a.y = xv * atB;
      return a;
    }, Wsc1, 256, lane, c);
  }

  const float scale = (variant < 4) ? 0.125f /*1/sqrt(64)*/ : 0.0625f /*1/sqrt(64*4)*/;
  const int ncol  = lane & 15;
  const int rbase = tile * 16 + (lane >> 4) * 8;
#pragma unroll
  for (int i = 0; i < 8; ++i) {
    const int row = rbase + i;
    if (row >= N) continue;
#pragma unroll
    for (int nt = 0; nt < 4; ++nt) {
      const int   col = nt * 16 + ncol;
      const float v   = c[nt][i] * scale;
      if (variant == 0) {
        y0[(size_t)row * 64 + col] = v;
      } else if (variant <= 3) {
        y1[(size_t)row * 192 + col * 3 + (variant - 1)] = v;
      } else if (variant == 4) {
        out[(size_t)row * 256 + col] = v;
      } else {
        out[(size_t)row * 256 + 64 + col * 3 + (variant - 5)] = v;
      }
    }
  }
}

// ---------------------------------------------------------------------------
// Edge kernel: persistent waves, grid-stride over edges.  Edge-invariant
// weights (Wfc1 column for lanes 0-7; all 64 Wfc2 values a lane ever needs,
// pre-scaled by 1/sqrt(H)*1/sqrt(32) and additionally 1/sqrt(3) for wp4) are
// hoisted into registers once.  Per edge: lanes 0-7 evaluate
// h = ssp(emb @ W_fc1 / sqrt(8)) and broadcast via shfl; each lane handles
// channels u = lane and lane+32, gathers y0/y1[src] (L2-resident) and
// scatter-adds 16 values into agg[dst] with hardware f32 atomics.
// ---------------------------------------------------------------------------
__global__ void edge_kernel(const float* __restrict__ edge_attrs,
                            const float* __restrict__ emb,
                            const int*   __restrict__ eidx,
                            const float* __restrict__ Wfc1,
                            const float* __restrict__ Wfc2,
                            const float* __restrict__ y0,
                            const float* __restrict__ y1,
                            float* __restrict__ agg,
                            int E, int nWaves) {
  const int lane = threadIdx.x & 31;
  const int wid  = blockIdx.x * (blockDim.x >> 5) + (threadIdx.x >> 5);

  const float inv_sqrt8 = 0.35355339059327373f;   // 1/sqrt(NB) == 1/sqrt(H)
  const float agg_scale = 0.17677669529663687f;   // 1/sqrt(AVG_NEIGH=32)
  const float inv_sqrt3 = 0.5773502691896258f;
  const float LOG2f     = 0.6931471805599453f;
  const float cs        = inv_sqrt8 * agg_scale;  // folded into Wfc2 regs

  // ---- hoisted, edge-invariant weights ----
  float wf1[8];
#pragma unroll
  for (int k = 0; k < 8; ++k) wf1[k] = (lane < 8) ? Wfc1[k * 8 + lane] : 0.0f;

  float wf2[4][2][8];   // [part][half][j], pre-scaled
#pragma unroll
  for (int part = 0; part < 4; ++part) {
    const float f = (part == 3) ? cs * inv_sqrt3 : cs;
#pragma unroll
    for (int half = 0; half < 2; ++half)
#pragma unroll
      for (int j = 0; j < 8; ++j)
        wf2[part][half][j] = Wfc2[j * 256 + part * 64 + half * 32 + lane] * f;
  }

  for (int e = wid; e < E; e += nWaves) {
    // h[j] computed by lane j (j < 8), then broadcast
    float hj = 0.0f;
    if (lane < 8) {
      float acc = 0.0f;
#pragma unroll
      for (int k = 0; k < 8; ++k) acc += emb[(size_t)e * 8 + k] * wf1[k];
      acc *= inv_sqrt8;
      // stable softplus, shifted: ssp(x) = softplus(x) - log(2)
      hj = fmaxf(acc, 0.0f) + log1pf(expf(-fabsf(acc))) - LOG2f;
    }
    float h[8];
#pragma unroll
    for (int j = 0; j < 8; ++j) h[j] = __shfl(hj, j, 32);

    const int   dst = eidx[e];
    const int   src = eidx[E + e];
    const float sh0 = edge_attrs[(size_t)e * 4 + 0];
    const float s1x = edge_attrs[(size_t)e * 4 + 1];
    const float s1y = edge_attrs[(size_t)e * 4 + 2];
    const float s1z = edge_attrs[(size_t)e * 4 + 3];

    float* const base = agg + (size_t)dst * 512;

#pragma unroll
    for (int half = 0; half < 2; ++half) {
      const int u = lane + 32 * half;
      float w1 = 0.f, w2 = 0.f, w3 = 0.f, w4 = 0.f;
#pragma unroll
      for (int j = 0; j < 8; ++j) {
        w1 += h[j] * wf2[0][half][j];
        w2 += h[j] * wf2[1][half][j];
        w3 += h[j] * wf2[2][half][j];
        w4 += h[j] * wf2[3][half][j];
      }

      const float xs0 = y0[(size_t)src * 64 + u];
      const float x1a = y1[(size_t)src * 192 + u * 3 + 0];
      const float x1b = y1[(size_t)src * 192 + u * 3 + 1];
      const float x1c = y1[(size_t)src * 192 + u * 3 + 2];
      const float dt  = x1a * s1x + x1b * s1y + x1c * s1z;

      const float ms0 = w1 * xs0 * sh0;   // scales already folded into w*
      const float ms1 = w4 * dt;
      const float wv2 = w2 * xs0;         // * sh1[m]
      const float wv3 = w3 * sh0;         // * xs1[m]

      unsafeAtomicAdd(base + u,               ms0);
      unsafeAtomicAdd(base + 64 + u,          ms1);
      unsafeAtomicAdd(base + 128 + u * 3 + 0, wv2 * s1x);
      unsafeAtomicAdd(base + 128 + u * 3 + 1, wv2 * s1y);
      unsafeAtomicAdd(base + 128 + u * 3 + 2, wv2 * s1z);
      unsafeAtomicAdd(base + 320 + u * 3 + 0, wv3 * x1a);
      unsafeAtomicAdd(base + 320 + u * 3 + 1, wv3 * x1b);
      unsafeAtomicAdd(base + 320 + u * 3 + 2, wv3 * x1c);
    }
  }
}

// ---------------------------------------------------------------------------
// Finalize: gridDim.y selects
//   0    : o0    = agg[:, :128]     @ W_lin2_0 / sqrt(128)   (+= into out)
//   1..3 : o1[m] = agg-vec-slice(m) @ W_lin2_1 / sqrt(128)   (+= into out)
// `out` already holds sc0/sc1 from the node kernel.
// ---------------------------------------------------------------------------
__global__ void finalize_kernel(const float* __restrict__ agg,
                                const float* __restrict__ Wl20,
                                const float* __restrict__ Wl21,
                                float* __restrict__ out,
                                int N, int nTiles) {
  const int lane = threadIdx.x & 31;
  const int wave = threadIdx.x >> 5;
  const int tile = blockIdx.x * (blockDim.x >> 5) + wave;
  if (tile >= nTiles) return;
  const int variant = blockIdx.y;

  int arow = tile * 16 + (lane & 15);
  if (arow >= N) arow = N - 1;
  const size_t an = (size_t)arow;

  v8f c[4] = {};

  if (variant == 0) {
    run_gemm16x64([&](int kb, int k0) {
      return *(const v2f*)(agg + an * 512 + k0);   // contiguous pair
    }, Wl20, 128, lane, c);
  } else {
    const int m = variant - 1;
    run_gemm16x64([&](int kb, int k0) {
      v2f a;
      a.x = agg[an * 512 + 128 + k0 * 3 + m];
      a.y = agg[an * 512 + 128 + (k0 + 1) * 3 + m];
      return a;
    }, Wl21, 128, lane, c);
  }

  const float scale = 0.08838834764831845f;  // 1/sqrt(2*C=128)
  const int ncol  = lane & 15;
  const int rbase = tile * 16 + (lane >> 4) * 8;
#pragma unroll
  for (int i = 0; i < 8; ++i) {
    const int row = rbase + i;
    if (row >= N) continue;
#pragma unroll
    for (int nt = 0; nt < 4; ++nt) {
      const int   col = nt * 16 + ncol;
      const float v   = c[nt][i] * scale;
      if (variant == 0) {
        out[(size_t)row * 256 + col] += v;
      } else {
        out[(size_t)row * 256 + 64 + col * 3 + (variant - 1)] += v;
      }
    }
  }
}

// ---------------------------------------------------------------------------
extern "C" void kernel_launch(void* const* d_in, const int* in_sizes, int n_in,
                              void* d_out, int out_size, void* d_ws, size_t ws_size,
                              hipStream_t stream) {
  const float* nf    = (const float*)d_in[0];   // (N, 256)
  const float* attrs = (const float*)d_in[1];   // (N, 4)
  const float* eatt  = (const float*)d_in[2];   // (E, 4)
  const float* eemb  = (const float*)d_in[3];   // (E, 8)
  const int*   eidx  = (const int*)  d_in[4];   // (2, E)
  const float* Wl10  = (const float*)d_in[5];   // (64, 64)
  const float* Wl11  = (const float*)d_in[6];   // (64, 64)
  const float* Wfc1  = (const float*)d_in[7];   // (8, 8)
  const float* Wfc2  = (const float*)d_in[8];   // (8, 256)
  const float* Wl20  = (const float*)d_in[9];   // (128, 64)
  const float* Wl21  = (const float*)d_in[10];  // (128, 64)
  const float* Wsc0  = (const float*)d_in[11];  // (64, 4, 64) -> flat (256, 64)
  const float* Wsc1  = (const float*)d_in[12];  // (64, 4, 64) -> flat (256, 64)
  float*       out   = (float*)d_out;

  const int N = in_sizes[0] / 256;
  const int E = in_sizes[2] / 4;
  const int nTiles = (N + 15) / 16;

  // workspace: agg (N x 512) | y0 (N x 64) | y1 (N x 192)
  float* agg = (float*)d_ws;
  float* y0  = agg + (size_t)N * 512;
  float* y1  = y0 + (size_t)N * 64;

  hipMemsetAsync(agg, 0, (size_t)N * 512 * sizeof(float), stream);

  {  // y0, y1, sc0, sc1 (sc written straight into d_out)
    dim3 grid((nTiles + 3) / 4, 8);
    node_gemm_kernel<<<grid, 128, 0, stream>>>(nf, attrs, Wl10, Wl11, Wsc0, Wsc1,
                                               y0, y1, out, N, nTiles);
  }
  {  // fused edge MLP + gather + scatter-add (persistent waves)
    const int blocks = 2048;                 // 8192 waves
    const int nWaves = blocks * 4;
    edge_kernel<<<blocks, 128, 0, stream>>>(eatt, eemb, eidx, Wfc1, Wfc2,
                                            y0, y1, agg, E, nWaves);
  }
  {  // o0 / o1 accumulated onto sc
    dim3 grid((nTiles + 3) / 4, 4);
    finalize_kernel<<<grid, 128, 0, stream>>>(agg, Wl20, Wl21, out, N, nTiles);
  }
}